// PointTransformerSeg_40054865002795
// MI455X (gfx1250) — compile-verified
//
#include <hip/hip_runtime.h>
#include <math.h>

// =====================================================================
// PointTransformerSeg forward for MI455X (gfx1250).
//  - All Linear layers use V_WMMA_F32_16X16X4_F32 (f32 matches reference).
//    Each wave computes a 16x32 C tile (2 accumulators sharing one A
//    fragment); K loop is split into a guard-free main body (K & ~3)
//    plus one guarded tail so the hot loop has no exec-masked loads.
//  - FPS: one workgroup per batch, serial rounds with LDS tree argmax.
//  - KNN (k=16 grouping, k=3 interpolation): brute force, reference
//    points tiled through LDS, top-k kept in registers per query.
//  - BatchNorm (train mode): exact per-channel mean/var over all rows
//    (one block per channel), then fused normalize+ReLU.
// Working set ~80MB -> L2 resident (192MB); HBM irrelevant.
// =====================================================================

typedef float v2f __attribute__((ext_vector_type(2)));
typedef float v8f __attribute__((ext_vector_type(8)));

static constexpr int BB      = 2;
static constexpr int NSL[5]  = {16384, 4096, 1024, 256, 64};
static constexpr int PLL[5]  = {32, 64, 128, 256, 512};
static constexpr int NSAMP   = 16;

// ---------------------------------------------------------------
// GEMM  C[M,N] = A[M,K] @ W[K,N] (+ bias), row-major f32.
// One wave -> 16x32 C tile via two V_WMMA_F32_16X16X4_F32 chains.
// Requirements (hold at every call site): M % 16 == 0, N % 32 == 0.
// K arbitrary (guarded tail handles K % 4).
// A 16x4 f32 fragment (ISA 7.12.2): lane = M (mod 16), half=lane/16
//   selects K pair {0,1} vs {2,3}; VGPR j holds K = 2*half + j.
// B 4x16 fragment symmetric (lane = N). C/D: VGPR v -> M = v + 8*half.
// ---------------------------------------------------------------
__global__ __launch_bounds__(256) void gemm_bias_wmma(
    const float* __restrict__ A, const float* __restrict__ W,
    const float* __restrict__ bias, float* __restrict__ C,
    int M, int N, int K)
{
  const int wave = threadIdx.x >> 5;
  const int lane = threadIdx.x & 31;
  const int tiles_n = N >> 5;                  // 32-wide super-tiles
  const int tiles   = (M >> 4) * tiles_n;
  const int tile    = blockIdx.x * 8 + wave;
  if (tile >= tiles) return;                   // wave-uniform exit
  const int tm = tile / tiles_n;
  const int tn = tile - tm * tiles_n;
  const int half = lane >> 4;
  const int l16  = lane & 15;
  const int row  = tm * 16 + l16;              // A row for this lane
  const int col0 = tn * 32 + l16;              // first 16-col tile
  const int col1 = col0 + 16;                  // second 16-col tile
  const float* __restrict__ Arow = A + (size_t)row * K;

  v8f acc0 = {0.f, 0.f, 0.f, 0.f, 0.f, 0.f, 0.f, 0.f};
  v8f acc1 = {0.f, 0.f, 0.f, 0.f, 0.f, 0.f, 0.f, 0.f};

  const int K4 = K & ~3;
  int k0 = 0;
#pragma unroll 2
  for (; k0 < K4; k0 += 4) {
    const int ka = k0 + 2 * half;
    v2f a, b0, b1;
    a.x  = Arow[ka];
    a.y  = Arow[ka + 1];
    const float* __restrict__ w0 = W + (size_t)ka * N;
    b0.x = w0[col0];
    b0.y = w0[N + col0];
    b1.x = w0[col1];
    b1.y = w0[N + col1];
    acc0 = __builtin_amdgcn_wmma_f32_16x16x4_f32(false, a, false, b0,
                                                 (short)0, acc0, false, false);
    acc1 = __builtin_amdgcn_wmma_f32_16x16x4_f32(false, a, false, b1,
                                                 (short)0, acc1, false, false);
  }
  if (k0 < K) {                                // K % 4 tail (guarded)
    const int ka = k0 + 2 * half;
    v2f a, b0, b1;
    const bool e0 = (ka     < K);
    const bool e1 = (ka + 1 < K);
    a.x  = e0 ? Arow[ka]                      : 0.f;
    a.y  = e1 ? Arow[ka + 1]                  : 0.f;
    b0.x = e0 ? W[(size_t)ka * N + col0]       : 0.f;
    b0.y = e1 ? W[(size_t)(ka + 1) * N + col0] : 0.f;
    b1.x = e0 ? W[(size_t)ka * N + col1]       : 0.f;
    b1.y = e1 ? W[(size_t)(ka + 1) * N + col1] : 0.f;
    acc0 = __builtin_amdgcn_wmma_f32_16x16x4_f32(false, a, false, b0,
                                                 (short)0, acc0, false, false);
    acc1 = __builtin_amdgcn_wmma_f32_16x16x4_f32(false, a, false, b1,
                                                 (short)0, acc1, false, false);
  }

  const float bv0 = bias ? bias[col0] : 0.f;
  const float bv1 = bias ? bias[col1] : 0.f;
#pragma unroll
  for (int v = 0; v < 8; ++v) {
    const int m = tm * 16 + v + 8 * half;
    C[(size_t)m * N + col0] = acc0[v] + bv0;
    C[(size_t)m * N + col1] = acc1[v] + bv1;
  }
}

// ---------------------------------------------------------------
// Furthest Point Sampling: one workgroup per batch (serial rounds).
// ---------------------------------------------------------------
__global__ __launch_bounds__(1024) void fps_kernel(
    const float* __restrict__ p, int n, int m,
    int* __restrict__ idx, float* __restrict__ distws)
{
  __shared__ float sv[1024];
  __shared__ int   si[1024];
  const int b   = blockIdx.x;
  const int tid = threadIdx.x;
  const int bd  = blockDim.x;
  const float* pb  = p + (size_t)b * n * 3;
  int*   ib   = idx    + (size_t)b * m;
  float* dist = distws + (size_t)b * n;
  for (int j = tid; j < n; j += bd) dist[j] = 1e10f;
  __syncthreads();
  int last = 0;
  for (int it = 0; it < m; ++it) {
    if (tid == 0) ib[it] = last;
    const float lx = pb[last * 3 + 0];
    const float ly = pb[last * 3 + 1];
    const float lz = pb[last * 3 + 2];
    float best = -1.f; int bj = 0;
    for (int j = tid; j < n; j += bd) {
      const float dx = pb[j * 3 + 0] - lx;
      const float dy = pb[j * 3 + 1] - ly;
      const float dz = pb[j * 3 + 2] - lz;
      const float d  = dx * dx + dy * dy + dz * dz;
      const float dm = fminf(dist[j], d);
      dist[j] = dm;
      if (dm > best) { best = dm; bj = j; }
    }
    sv[tid] = best; si[tid] = bj;
    __syncthreads();
    for (int s = bd >> 1; s > 0; s >>= 1) {
      if (tid < s) {
        if (sv[tid + s] > sv[tid] ||
            (sv[tid + s] == sv[tid] && si[tid + s] < si[tid])) {
          sv[tid] = sv[tid + s]; si[tid] = si[tid + s];
        }
      }
      __syncthreads();
    }
    last = si[0];
    __syncthreads();
  }
}

// out[b,i,:] = p[b, idx[b,i], :]
__global__ void gather_points_kernel(
    const float* __restrict__ p, const int* __restrict__ idx,
    float* __restrict__ out, int n, int m)
{
  const size_t total = (size_t)BB * m * 3;
  const size_t i = (size_t)blockIdx.x * blockDim.x + threadIdx.x;
  if (i >= total) return;
  const int c  = (int)(i % 3);
  const size_t r = i / 3;                 // b*m + q
  const int b  = (int)(r / m);
  out[i] = p[((size_t)b * n + idx[r]) * 3 + c];
}

// ---------------------------------------------------------------
// Brute-force KNN: one thread per query, refs tiled through LDS,
// top-K kept sorted in registers (stable -> matches top_k ties).
// ---------------------------------------------------------------
template <int KNN>
__global__ __launch_bounds__(256) void knn_kernel(
    const float* __restrict__ q, const float* __restrict__ r,
    int mq, int nr, int* __restrict__ oidx, float* __restrict__ odist)
{
  constexpr int TILE = 256;
  __shared__ float sx[TILE], sy[TILE], sz[TILE];
  const int nblk = (mq + 255) >> 8;
  const int b    = blockIdx.x / nblk;
  const int qi   = (blockIdx.x % nblk) * 256 + threadIdx.x;
  const bool valid = qi < mq;
  const float* qb = q + (size_t)b * mq * 3;
  const float* rb = r + (size_t)b * nr * 3;
  float qx = 0.f, qy = 0.f, qz = 0.f;
  if (valid) { qx = qb[qi*3]; qy = qb[qi*3+1]; qz = qb[qi*3+2]; }
  float bd[KNN]; int bi[KNN];
#pragma unroll
  for (int t = 0; t < KNN; ++t) { bd[t] = 3.4e38f; bi[t] = 0; }
  for (int base = 0; base < nr; base += TILE) {
    const int cnt = (nr - base < TILE) ? (nr - base) : TILE;
    for (int t = threadIdx.x; t < cnt; t += 256) {
      sx[t] = rb[(base + t) * 3 + 0];
      sy[t] = rb[(base + t) * 3 + 1];
      sz[t] = rb[(base + t) * 3 + 2];
    }
    __syncthreads();
    if (valid) {
      for (int t = 0; t < cnt; ++t) {
        const float dx = sx[t] - qx, dy = sy[t] - qy, dz = sz[t] - qz;
        const float d = dx * dx + dy * dy + dz * dz;
        if (d < bd[KNN - 1]) {
          int pos = KNN - 1;
          while (pos > 0 && bd[pos - 1] > d) {
            bd[pos] = bd[pos - 1]; bi[pos] = bi[pos - 1]; --pos;
          }
          bd[pos] = d; bi[pos] = base + t;
        }
      }
    }
    __syncthreads();
  }
  if (valid) {
    const size_t o = ((size_t)b * mq + qi) * KNN;
#pragma unroll
    for (int t = 0; t < KNN; ++t) {
      oidx[o + t] = bi[t];
      if (odist) odist[o + t] = bd[t];
    }
  }
}

// gx rows [(b*m+q)*16+s, 3+cp]: rel xyz then gathered features
__global__ void build_gx_kernel(
    const float* __restrict__ pprev, const float* __restrict__ xprev,
    const float* __restrict__ pq, const int* __restrict__ nidx,
    float* __restrict__ A, int np, int m, int cp)
{
  const int K = 3 + cp;
  const size_t total = (size_t)BB * m * NSAMP * K;
  const size_t i = (size_t)blockIdx.x * blockDim.x + threadIdx.x;
  if (i >= total) return;
  const int c = (int)(i % K);
  const size_t r = i / K;                 // (b*m+q)*16 + s
  const int s = (int)(r % NSAMP);
  const size_t bq = r / NSAMP;            // b*m + q
  const int b = (int)(bq / m);
  const int j = nidx[bq * NSAMP + s];
  float v;
  if (c < 3) v = pprev[((size_t)b * np + j) * 3 + c] - pq[bq * 3 + c];
  else       v = xprev[((size_t)b * np + j) * (size_t)cp + (c - 3)];
  A[i] = v;
}

// per-channel train-mode BN stats: one block per channel
__global__ __launch_bounds__(256) void bn_stats_kernel(
    const float* __restrict__ x, int M, int C,
    float* __restrict__ mean, float* __restrict__ rstd)
{
  __shared__ float ss[256], s2[256];
  const int c = blockIdx.x;
  float a = 0.f, b = 0.f;
  for (int r = threadIdx.x; r < M; r += 256) {
    const float v = x[(size_t)r * C + c];
    a += v; b += v * v;
  }
  ss[threadIdx.x] = a; s2[threadIdx.x] = b;
  __syncthreads();
  for (int st = 128; st > 0; st >>= 1) {
    if (threadIdx.x < st) {
      ss[threadIdx.x] += ss[threadIdx.x + st];
      s2[threadIdx.x] += s2[threadIdx.x + st];
    }
    __syncthreads();
  }
  if (threadIdx.x == 0) {
    const float mu  = ss[0] / (float)M;
    const float var = fmaxf(s2[0] / (float)M - mu * mu, 0.f);
    mean[c] = mu;
    rstd[c] = rsqrtf(var + 1e-5f);
  }
}

__global__ void bn_relu_kernel(
    const float* __restrict__ x, float* __restrict__ y, int M, int C,
    const float* __restrict__ mean, const float* __restrict__ rstd,
    const float* __restrict__ g, const float* __restrict__ b)
{
  const size_t total = (size_t)M * C;
  const size_t i = (size_t)blockIdx.x * blockDim.x + threadIdx.x;
  if (i >= total) return;
  const int c = (int)(i % C);
  const float v = (x[i] - mean[c]) * rstd[c] * g[c] + b[c];
  y[i] = v > 0.f ? v : 0.f;
}

// max over the 16-neighbor group
__global__ void group_max_kernel(
    const float* __restrict__ pre, float* __restrict__ out, int rows, int C)
{
  const size_t total = (size_t)rows * C;
  const size_t i = (size_t)blockIdx.x * blockDim.x + threadIdx.x;
  if (i >= total) return;
  const int c = (int)(i % C);
  const size_t r = i / C;
  float mx = -3.4e38f;
#pragma unroll
  for (int s = 0; s < NSAMP; ++s)
    mx = fmaxf(mx, pre[((size_t)r * NSAMP + s) * C + c]);
  out[i] = mx;
}

// mean over n rows per (b,c)
__global__ void rowmean_kernel(
    const float* __restrict__ x, float* __restrict__ out, int n, int C)
{
  const int i = blockIdx.x * blockDim.x + threadIdx.x;
  if (i >= BB * C) return;
  const int b = i / C, c = i % C;
  float s = 0.f;
  for (int r = 0; r < n; ++r) s += x[((size_t)b * n + r) * C + c];
  out[i] = s / (float)n;
}

// out[b,o] = relu(in[b,:] @ W + bias) (tiny: 2x512)
__global__ void glob_linear_kernel(
    const float* __restrict__ in, const float* __restrict__ W,
    const float* __restrict__ bias, float* __restrict__ out, int Ci, int Co)
{
  const int i = blockIdx.x * blockDim.x + threadIdx.x;
  if (i >= BB * Co) return;
  const int b = i / Co, o = i % Co;
  float s = bias[o];
  for (int c = 0; c < Ci; ++c) s += in[b * Ci + c] * W[(size_t)c * Co + o];
  out[i] = s > 0.f ? s : 0.f;
}

// xc[b,r,:] = [x5[b,r,:512], glob[b,:512]]
__global__ void concat_xc_kernel(
    const float* __restrict__ x5, const float* __restrict__ glob,
    float* __restrict__ xc)
{
  const size_t total = (size_t)BB * 64 * 1024;
  const size_t i = (size_t)blockIdx.x * blockDim.x + threadIdx.x;
  if (i >= total) return;
  const int c = (int)(i % 1024);
  const size_t r = i / 1024;              // b*64 + row
  const int b = (int)(r / 64);
  xc[i] = (c < 512) ? x5[r * 512 + c] : glob[b * 512 + (c - 512)];
}

// out = a + 3-NN inverse-distance interpolation of cfeat
__global__ void interp_add_kernel(
    const float* __restrict__ a, const float* __restrict__ cfeat,
    const int* __restrict__ iidx, const float* __restrict__ sqd,
    float* __restrict__ out, int nf, int nc, int C)
{
  const size_t total = (size_t)BB * nf * C;
  const size_t i = (size_t)blockIdx.x * blockDim.x + threadIdx.x;
  if (i >= total) return;
  const int ch = (int)(i % C);
  const size_t r = i / C;                 // b*nf + fi
  const int b = (int)(r / nf);
  float wsum = 0.f, s = 0.f;
#pragma unroll
  for (int j = 0; j < 3; ++j) {
    const float d = sqrtf(sqd[r * 3 + j]);
    const float w = 1.f / (d + 1e-8f);
    wsum += w;
    const int ci = iidx[r * 3 + j];
    s += w * cfeat[((size_t)b * nc + ci) * C + ch];
  }
  out[i] = a[i] + s / wsum;
}

// A6[r,:] = [p0[r,:3], x0[r,:3]]
__global__ void build_enc1_kernel(
    const float* __restrict__ p0, const float* __restrict__ x0,
    float* __restrict__ A)
{
  const size_t total = (size_t)BB * 16384 * 6;
  const size_t i = (size_t)blockIdx.x * blockDim.x + threadIdx.x;
  if (i >= total) return;
  const int c = (int)(i % 6);
  const size_t r = i / 6;
  A[i] = (c < 3) ? p0[r * 3 + c] : x0[r * 3 + (c - 3)];
}

// =====================================================================
static inline unsigned gblk(size_t n) { return (unsigned)((n + 255) / 256); }

extern "C" void kernel_launch(void* const* d_in, const int* in_sizes, int n_in,
                              void* d_out, int out_size, void* d_ws, size_t ws_size,
                              hipStream_t stream)
{
  (void)in_sizes; (void)n_in; (void)out_size; (void)ws_size;
  const float* p0 = (const float*)d_in[0];
  const float* x0 = (const float*)d_in[1];
  auto P = [&](int i) { return (const float*)d_in[i]; };

  float* ws = (float*)d_ws;
  size_t off = 0;
  auto alloc = [&](size_t n) { size_t o = off; off += n; return o; };
  const size_t N0 = 16384, N1 = 4096, N2 = 1024, N3 = 256, N4 = 64;
  // persistent level features / positions
  float* xs[5];
  xs[0] = ws + alloc((size_t)BB * N0 * 32);
  xs[1] = ws + alloc((size_t)BB * N1 * 64);
  xs[2] = ws + alloc((size_t)BB * N2 * 128);
  xs[3] = ws + alloc((size_t)BB * N3 * 256);
  xs[4] = ws + alloc((size_t)BB * N4 * 512);
  float* ps_ws[5] = {nullptr, nullptr, nullptr, nullptr, nullptr};
  ps_ws[1] = ws + alloc((size_t)BB * N1 * 3);
  ps_ws[2] = ws + alloc((size_t)BB * N2 * 3);
  ps_ws[3] = ws + alloc((size_t)BB * N3 * 3);
  ps_ws[4] = ws + alloc((size_t)BB * N4 * 3);
  const float* ps[5] = {p0, ps_ws[1], ps_ws[2], ps_ws[3], ps_ws[4]};
  // scratch
  float* gx   = ws + alloc(4587520);            // max A matrix (enc2: 131072x35)
  float* pre  = ws + alloc(8388608);            // max pre-activation (enc2: 131072x64)
  float* abuf = ws + alloc((size_t)BB * N0 * 32); // decoder skip branch
  float* xc5  = ws + alloc((size_t)BB * N4 * 512);
  float* xc4  = ws + alloc((size_t)BB * N3 * 256);
  float* xc3  = ws + alloc((size_t)BB * N2 * 128);
  float* xc2  = ws + alloc((size_t)BB * N1 * 64);
  float* mean = ws + alloc(512);
  float* rstd = ws + alloc(512);
  float* gm   = ws + alloc((size_t)BB * 512);
  float* go   = ws + alloc((size_t)BB * 512);
  float* fdist= ws + alloc((size_t)BB * N0);
  float* iw   = ws + alloc((size_t)BB * N0 * 3); // knn sq-dists for interp
  int*   nidx = (int*)(ws + alloc(131072));      // max(B*4096*16, B*16384*3)
  int*   fidx = (int*)(ws + alloc((size_t)BB * N1));

  auto gemm = [&](const float* A, const float* W, const float* bias,
                  float* C, int M, int N, int K) {
    const unsigned blocks = (unsigned)(((size_t)(M / 16) * (N / 32) + 7) / 8);
    gemm_bias_wmma<<<blocks, 256, 0, stream>>>(A, W, bias, C, M, N, K);
  };

  // ---- enc1: concat(p, feat) -> Linear(6,32) -> BN -> ReLU ----
  build_enc1_kernel<<<gblk((size_t)BB * N0 * 6), 256, 0, stream>>>(p0, x0, gx);
  gemm(gx, P(3), nullptr, pre, BB * (int)N0, 32, 6);
  bn_stats_kernel<<<32, 256, 0, stream>>>(pre, BB * (int)N0, 32, mean, rstd);
  bn_relu_kernel<<<gblk((size_t)BB * N0 * 32), 256, 0, stream>>>(
      pre, xs[0], BB * (int)N0, 32, mean, rstd, P(4), P(5));

  // ---- enc2..enc5 (TransitionDown, stride 4) ----
  for (int l = 0; l < 4; ++l) {
    const int np = NSL[l], m = NSL[l + 1];
    const int cp = PLL[l], co = PLL[l + 1];
    const int wi = 6 + l * 3;   // enc{l+2}_w, _g, _b
    fps_kernel<<<BB, 1024, 0, stream>>>(ps[l], np, m, fidx, fdist);
    gather_points_kernel<<<gblk((size_t)BB * m * 3), 256, 0, stream>>>(
        ps[l], fidx, ps_ws[l + 1], np, m);
    knn_kernel<16><<<BB * ((m + 255) / 256), 256, 0, stream>>>(
        ps[l + 1], ps[l], m, np, nidx, nullptr);
    const int K = 3 + cp;
    build_gx_kernel<<<gblk((size_t)BB * m * NSAMP * K), 256, 0, stream>>>(
        ps[l], xs[l], ps[l + 1], nidx, gx, np, m, cp);
    const int M = BB * m * NSAMP;
    gemm(gx, P(wi), nullptr, pre, M, co, K);
    bn_stats_kernel<<<co, 256, 0, stream>>>(pre, M, co, mean, rstd);
    bn_relu_kernel<<<gblk((size_t)M * co), 256, 0, stream>>>(
        pre, pre, M, co, mean, rstd, P(wi + 1), P(wi + 2));
    group_max_kernel<<<gblk((size_t)BB * m * co), 256, 0, stream>>>(
        pre, xs[l + 1], BB * m, co);
  }

  // ---- dec5 head ----
  rowmean_kernel<<<gblk((size_t)BB * 512), 256, 0, stream>>>(xs[4], gm, 64, 512);
  glob_linear_kernel<<<gblk((size_t)BB * 512), 256, 0, stream>>>(
      gm, P(22), P(23), go, 512, 512);
  concat_xc_kernel<<<gblk((size_t)BB * 64 * 1024), 256, 0, stream>>>(xs[4], go, gx);
  gemm(gx, P(18), P(19), pre, BB * 64, 512, 1024);
  bn_stats_kernel<<<512, 256, 0, stream>>>(pre, BB * 64, 512, mean, rstd);
  bn_relu_kernel<<<gblk((size_t)BB * 64 * 512), 256, 0, stream>>>(
      pre, xc5, BB * 64, 512, mean, rstd, P(20), P(21));

  // ---- dec4..dec1 ----
  float* xcur = xc5;
  float* xcbuf[4] = {xc4, xc3, xc2, (float*)d_out};
  for (int l = 4; l >= 1; --l) {
    const int nf = NSL[l - 1], nc = NSL[l];
    const int co = PLL[l - 1], ci = PLL[l];
    const int base = 24 + (4 - l) * 8;
    // skip branch: a = relu(bn(xs[l-1] @ l1_w + l1_b))
    {
      const int M = BB * nf;
      gemm(xs[l - 1], P(base + 0), P(base + 1), pre, M, co, co);
      bn_stats_kernel<<<co, 256, 0, stream>>>(pre, M, co, mean, rstd);
      bn_relu_kernel<<<gblk((size_t)M * co), 256, 0, stream>>>(
          pre, abuf, M, co, mean, rstd, P(base + 2), P(base + 3));
    }
    // coarse branch: c = relu(bn(xcur @ l2_w + l2_b))
    {
      const int M = BB * nc;
      gemm(xcur, P(base + 4), P(base + 5), pre, M, co, ci);
      bn_stats_kernel<<<co, 256, 0, stream>>>(pre, M, co, mean, rstd);
      bn_relu_kernel<<<gblk((size_t)M * co), 256, 0, stream>>>(
          pre, pre, M, co, mean, rstd, P(base + 6), P(base + 7));
    }
    // 3-NN of fine points among coarse, inverse-distance interpolate + add
    knn_kernel<3><<<BB * ((nf + 255) / 256), 256, 0, stream>>>(
        ps[l - 1], ps[l], nf, nc, nidx, iw);
    float* out = xcbuf[4 - l];
    interp_add_kernel<<<gblk((size_t)BB * nf * co), 256, 0, stream>>>(
        abuf, pre, nidx, iw, out, nf, nc, co);
    xcur = out;
  }
}